// WindowAttention_49100066128647
// MI455X (gfx1250) — compile-verified
//
#include <hip/hip_runtime.h>
#include <math.h>

typedef __attribute__((ext_vector_type(2))) float v2f;
typedef __attribute__((ext_vector_type(8))) float v8f;

#define DIMC 256
#define NHEAD 8
#define HD 32
#define NWIN 64
#define MTOK 50   // tokens (cls + 49 patches)
#define NQ 49     // query rows (cls dropped)

// One workgroup (4 wave32s) per (window, head).
__global__ __launch_bounds__(128) void win_attn_kernel(
    const float* __restrict__ qkv,        // (4096, 50, 768)
    const float* __restrict__ mask,       // (64, 49, 50)
    const float* __restrict__ bias_table, // (169, 8)
    const int*   __restrict__ rel_index,  // (49, 49)
    float* __restrict__ out)              // (4096, 49, 256)
{
    __shared__ float Qs[64 * 32];        // query rows, scaled, padded
    __shared__ float Ks[64 * 32];        // key rows, padded
    __shared__ float Vt[32 * 64];        // V transposed: Vt[d][tok], tok padded
    __shared__ float Bs[NQ * NQ];        // per-head relative-position bias
    __shared__ float Ps[4 * 16 * 64];    // per-wave softmax probs (A-layout source)

    const int bh   = blockIdx.x;
    const int b    = bh >> 3;
    const int h    = bh & 7;
    const int tid  = threadIdx.x;
    const int wave = tid >> 5;           // query tile 0..3
    const int lane = tid & 31;
    const int g    = lane >> 4;          // lane group (K-offset selector)
    const int lm   = lane & 15;

    const float scale = 0.17677669529663687f;  // 32^-0.5
    const float* base = qkv + (size_t)b * (MTOK * 768);

    // ---- stage Q (scaled; query m -> token m+1) and K into LDS, zero-padded
    for (int idx = tid; idx < 64 * 32; idx += 128) {
        int row = idx >> 5, d = idx & 31;
        float qv = 0.f, kv = 0.f;
        if (row < NQ)   qv = base[(row + 1) * 768 +       h * HD + d] * scale;
        if (row < MTOK) kv = base[ row      * 768 + 256 + h * HD + d];
        Qs[idx] = qv;
        Ks[idx] = kv;
    }
    // ---- stage V transposed: Vt[d][t], token axis padded to 64 with zeros
    for (int idx = tid; idx < 32 * 64; idx += 128) {
        int d = idx >> 6, t = idx & 63;
        Vt[idx] = (t < MTOK) ? base[t * 768 + 512 + h * HD + d] : 0.f;
    }
    // ---- stage relative-position bias for this head
    for (int idx = tid; idx < NQ * NQ; idx += 128) {
        Bs[idx] = bias_table[rel_index[idx] * NHEAD + h];
    }
    __syncthreads();

    // ---- GEMM1: S(16x64) = Q_tile(16x32) @ K^T(32x64), f32 WMMA K-step 4
    v2f a[8];
    #pragma unroll
    for (int c = 0; c < 8; ++c)
        a[c] = *(const v2f*)&Qs[(wave * 16 + lm) * 32 + 4 * c + 2 * g];

    v8f S[4];
    #pragma unroll
    for (int nt = 0; nt < 4; ++nt) {
        v8f acc = {};
        #pragma unroll
        for (int c = 0; c < 8; ++c) {
            v2f bf = *(const v2f*)&Ks[(nt * 16 + lm) * 32 + 4 * c + 2 * g];
            acc = __builtin_amdgcn_wmma_f32_16x16x4_f32(
                false, a[c], false, bf, (short)0, acc, false, false);
        }
        S[nt] = acc;
    }

    // ---- bias + window mask + padding fixup; track row max
    const float* mrow = mask + (size_t)(b & (NWIN - 1)) * (NQ * MTOK);
    float rmax[8], rsum[8];
    #pragma unroll
    for (int r = 0; r < 8; ++r) rmax[r] = -INFINITY;

    #pragma unroll
    for (int nt = 0; nt < 4; ++nt) {
        int n = nt * 16 + lm;
        #pragma unroll
        for (int r = 0; r < 8; ++r) {
            int mq = wave * 16 + r + 8 * g;
            float s = S[nt][r];
            if (n < MTOK && mq < NQ) {
                s += mrow[mq * MTOK + n];
                if (n >= 1) s += Bs[mq * NQ + (n - 1)];
            } else {
                s = -INFINITY;
            }
            S[nt][r] = s;
            rmax[r] = fmaxf(rmax[r], s);
        }
    }

    // ---- softmax: rows live across the 16 lanes of each group
    #pragma unroll
    for (int r = 0; r < 8; ++r) {
        float m = rmax[r];
        #pragma unroll
        for (int off = 1; off < 16; off <<= 1)
            m = fmaxf(m, __shfl_xor(m, off, 32));
        rmax[r] = m;
        rsum[r] = 0.f;
    }
    #pragma unroll
    for (int nt = 0; nt < 4; ++nt) {
        #pragma unroll
        for (int r = 0; r < 8; ++r) {
            float p = 0.f;
            if (rmax[r] > -INFINITY) p = __expf(S[nt][r] - rmax[r]);
            S[nt][r] = p;
            rsum[r] += p;
        }
    }
    #pragma unroll
    for (int r = 0; r < 8; ++r) {
        float s = rsum[r];
        #pragma unroll
        for (int off = 1; off < 16; off <<= 1)
            s += __shfl_xor(s, off, 32);
        rsum[r] = (s > 0.f) ? (1.0f / s) : 0.f;
    }

    // ---- write normalized probs to LDS (C-layout -> memory row-major)
    #pragma unroll
    for (int nt = 0; nt < 4; ++nt) {
        int n = nt * 16 + lm;
        #pragma unroll
        for (int r = 0; r < 8; ++r) {
            int ml = r + 8 * g;
            Ps[wave * 1024 + ml * 64 + n] = S[nt][r] * rsum[r];
        }
    }
    __syncthreads();

    // ---- GEMM2: D(16x32) = P(16x64) @ V(64x32), token axis is K
    #pragma unroll
    for (int ot = 0; ot < 2; ++ot) {
        v8f acc = {};
        #pragma unroll
        for (int c = 0; c < 16; ++c) {
            v2f pa = *(const v2f*)&Ps[wave * 1024 + lm * 64 + 4 * c + 2 * g];
            v2f vb = *(const v2f*)&Vt[(ot * 16 + lm) * 64 + 4 * c + 2 * g];
            acc = __builtin_amdgcn_wmma_f32_16x16x4_f32(
                false, pa, false, vb, (short)0, acc, false, false);
        }
        #pragma unroll
        for (int r = 0; r < 8; ++r) {
            int mq = wave * 16 + r + 8 * g;
            if (mq < NQ) {
                out[(size_t)b * (NQ * DIMC) + mq * DIMC + h * HD + ot * 16 + lm]
                    = acc[r];
            }
        }
    }
}

extern "C" void kernel_launch(void* const* d_in, const int* in_sizes, int n_in,
                              void* d_out, int out_size, void* d_ws, size_t ws_size,
                              hipStream_t stream) {
    (void)in_sizes; (void)n_in; (void)out_size; (void)d_ws; (void)ws_size;
    const float* qkv        = (const float*)d_in[0];
    const float* mask       = (const float*)d_in[1];
    const float* bias_table = (const float*)d_in[2];
    const int*   rel_index  = (const int*)d_in[3];
    float* out = (float*)d_out;

    dim3 grid(4096 * NHEAD);   // one workgroup per (window, head)
    dim3 block(128);           // 4 wave32s, one query tile each
    win_attn_kernel<<<grid, block, 0, stream>>>(qkv, mask, bias_table, rel_index, out);
}